// SketchPredictor_6579889897818
// MI455X (gfx1250) — compile-verified
//
#include <hip/hip_runtime.h>
#include <hip/hip_bf16.h>

typedef __attribute__((ext_vector_type(2))) float v2f;
typedef __attribute__((ext_vector_type(8))) float v8f;

constexpr int NN    = 1024;       // strokes per sketch
constexpr int ED    = 128;        // d_model
constexpr int BB    = 8;          // batch
constexpr int HH    = 8;          // heads
constexpr int LT    = 8;          // transformer layers
constexpr int LG    = 4;          // gcn layers
constexpr int NEDGE = 16 * NN;    // 16384 edges per sketch
constexpr int MROWS = BB * NN;    // 8192 flattened rows

__device__ __forceinline__ v8f wmma4(v2f a, v2f b, v8f c) {
  return __builtin_amdgcn_wmma_f32_16x16x4_f32(
      /*neg_a=*/false, a, /*neg_b=*/false, b,
      /*c_mod=*/(short)0, c, /*reuse_a=*/false, /*reuse_b=*/false);
}

// ---------------------------------------------------------------------------
// C[M,Nc] = A[M,K] @ W[K,Nc] + bias  (optional ReLU). One wave per 16x16 tile.
// ---------------------------------------------------------------------------
template <int RELU>
__global__ void gemm_bias_kernel(const float* __restrict__ A,
                                 const float* __restrict__ W,
                                 const float* __restrict__ bias,
                                 float* __restrict__ C,
                                 int K, int Nc) {
  const int lane = threadIdx.x & 31;
  const int l16  = lane & 15;
  const int half = lane >> 4;
  const int tm   = blockIdx.x * 16;
  const int tn   = blockIdx.y * 16;

  v8f acc = {};
  const float* arow = A + (size_t)(tm + l16) * K;
  for (int k0 = 0; k0 < K; k0 += 4) {
    v2f a, b;
    const float* ap = arow + k0 + 2 * half;
    a.x = ap[0];
    a.y = ap[1];
    const float* bp = W + (size_t)(k0 + 2 * half) * Nc + tn + l16;
    b.x = bp[0];
    b.y = bp[Nc];
    acc = wmma4(a, b, acc);
  }
  const float bv = bias ? bias[tn + l16] : 0.0f;
#pragma unroll
  for (int r = 0; r < 8; ++r) {
    float v = acc[r] + bv;
    if (RELU) v = fmaxf(v, 0.0f);
    C[(size_t)(tm + r + 8 * half) * Nc + tn + l16] = v;
  }
}

// ---------------------------------------------------------------------------
// Fused flash attention. qkv: [B, N, 3*ED] rows = [q | k | v]; head slice DH=16.
// grid (N/16, H, B), block 32 (one wave). Online softmax, P via LDS re-layout.
// ---------------------------------------------------------------------------
__global__ void attention_kernel(const float* __restrict__ qkv,
                                 float* __restrict__ o) {
  const int lane = threadIdx.x & 31;
  const int l16  = lane & 15;
  const int half = lane >> 4;
  const int qt   = blockIdx.x;
  const int h    = blockIdx.y;
  const int b    = blockIdx.z;
  const int ld   = 3 * ED;

  const float* base = qkv + (size_t)b * NN * ld;
  const float* Qm = base + h * 16;
  const float* Km = base + ED + h * 16;
  const float* Vm = base + 2 * ED + h * 16;

  // Preload Q A-fragments (16 rows x 16 dh, 4 K-steps of 4)
  v2f qa[4];
  const int qrow = qt * 16 + l16;
#pragma unroll
  for (int s = 0; s < 4; ++s) {
    const float* qp = Qm + (size_t)qrow * ld + s * 4 + 2 * half;
    qa[s].x = qp[0];
    qa[s].y = qp[1];
  }

  v8f oacc = {};
  float mrow[8], lrow[8];
#pragma unroll
  for (int r = 0; r < 8; ++r) { mrow[r] = -__builtin_inff(); lrow[r] = 0.0f; }

  __shared__ float sP[16][17];

  for (int kt = 0; kt < NN / 16; ++kt) {
    // S = Q @ K^T  (B[d][j] = K[j][d])
    v8f s = {};
#pragma unroll
    for (int st = 0; st < 4; ++st) {
      v2f bf;
      const float* kp = Km + (size_t)(kt * 16 + l16) * ld + st * 4 + 2 * half;
      bf.x = kp[0];
      bf.y = kp[1];
      s = wmma4(qa[st], bf, s);
    }
    // Online softmax per row (rows r+8*half live in 16-lane groups)
#pragma unroll
    for (int r = 0; r < 8; ++r) {
      float sv = s[r] * 0.25f;  // 1/sqrt(DH)
      float mx = sv;
#pragma unroll
      for (int off = 1; off < 16; off <<= 1) mx = fmaxf(mx, __shfl_xor(mx, off, 32));
      float mnew = fmaxf(mrow[r], mx);
      float p = __expf(sv - mnew);
      float ps = p;
#pragma unroll
      for (int off = 1; off < 16; off <<= 1) ps += __shfl_xor(ps, off, 32);
      float corr = __expf(mrow[r] - mnew);
      lrow[r] = lrow[r] * corr + ps;
      mrow[r] = mnew;
      oacc[r] *= corr;
      s[r] = p;
    }
    // C-layout -> LDS -> A-layout for P
#pragma unroll
    for (int r = 0; r < 8; ++r) sP[r + 8 * half][l16] = s[r];
    __syncthreads();
#pragma unroll
    for (int st = 0; st < 4; ++st) {
      v2f pa, vb;
      pa.x = sP[l16][st * 4 + 2 * half];
      pa.y = sP[l16][st * 4 + 2 * half + 1];
      const float* vp = Vm + (size_t)(kt * 16 + st * 4 + 2 * half) * ld + l16;
      vb.x = vp[0];
      vb.y = vp[ld];
      oacc = wmma4(pa, vb, oacc);
    }
    __syncthreads();
  }

  float* op = o + ((size_t)b * NN + qt * 16) * ED + h * 16;
#pragma unroll
  for (int r = 0; r < 8; ++r)
    op[(size_t)(r + 8 * half) * ED + l16] = oacc[r] / lrow[r];
}

// ---------------------------------------------------------------------------
// x = LayerNorm(x + t) * w + b ; one wave per 128-wide row.
// ---------------------------------------------------------------------------
__global__ void ln_residual_kernel(float* __restrict__ x,
                                   const float* __restrict__ t,
                                   const float* __restrict__ w,
                                   const float* __restrict__ bp) {
  const int row  = blockIdx.x;
  const int lane = threadIdx.x;
  float v[4];
  float s = 0.0f;
#pragma unroll
  for (int i = 0; i < 4; ++i) {
    const int c = lane * 4 + i;
    float r = x[(size_t)row * ED + c] + t[(size_t)row * ED + c];
    v[i] = r;
    s += r;
  }
#pragma unroll
  for (int off = 1; off < 32; off <<= 1) s += __shfl_xor(s, off, 32);
  const float mean = s * (1.0f / 128.0f);
  float var = 0.0f;
#pragma unroll
  for (int i = 0; i < 4; ++i) { float d = v[i] - mean; var += d * d; }
#pragma unroll
  for (int off = 1; off < 32; off <<= 1) var += __shfl_xor(var, off, 32);
  const float rs = rsqrtf(var * (1.0f / 128.0f) + 1e-5f);
#pragma unroll
  for (int i = 0; i < 4; ++i) {
    const int c = lane * 4 + i;
    x[(size_t)row * ED + c] = (v[i] - mean) * rs * w[c] + bp[c];
  }
}

// --------------------------- GCN helpers -----------------------------------
__global__ void zero_kernel(float* __restrict__ p, int n) {
  int i = blockIdx.x * 256 + threadIdx.x;
  if (i < n) p[i] = 0.0f;
}

__global__ void deg_kernel(const int* __restrict__ ei, float* __restrict__ deg) {
  int e = blockIdx.x * 256 + threadIdx.x;
  int b = blockIdx.y;
  if (e < NEDGE) {
    int col = ei[(size_t)b * 2 * NEDGE + NEDGE + e];
    atomicAdd(&deg[b * NN + col], 1.0f);
  }
}

__global__ void dinv_kernel(float* __restrict__ deg) {
  int i = blockIdx.x * 256 + threadIdx.x;  // B*N total
  deg[i] = rsqrtf(deg[i] + 1.0f);          // +1 self loop, always > 0
}

// out = dinv^2 * xw + bias  (self loop term; initializes the buffer)
__global__ void gcn_init_kernel(const float* __restrict__ dinv,
                                const float* __restrict__ xw,
                                const float* __restrict__ bias,
                                float* __restrict__ out) {
  int n = blockIdx.x, b = blockIdx.y, c = threadIdx.x;
  float di = dinv[b * NN + n];
  size_t idx = ((size_t)b * NN + n) * ED + c;
  out[idx] = di * di * xw[idx] + bias[c];
}

__global__ void gcn_scatter_kernel(const int* __restrict__ ei,
                                   const float* __restrict__ dinv,
                                   const float* __restrict__ xw,
                                   float* __restrict__ out) {
  int t = threadIdx.x;
  int e = blockIdx.x * 2 + (t >> 7);
  int c = t & 127;
  int b = blockIdx.y;
  if (e < NEDGE) {
    const int* eb = ei + (size_t)b * 2 * NEDGE;
    int row = eb[e];
    int col = eb[NEDGE + e];
    float nrm = dinv[b * NN + row] * dinv[b * NN + col];
    atomicAdd(&out[((size_t)b * NN + col) * ED + c],
              nrm * xw[((size_t)b * NN + row) * ED + c]);
  }
}

// ---------------------------------------------------------------------------
// out[row] = sigmoid([t_out, g_out] @ fc_W + fc_b); one wave per row.
// ---------------------------------------------------------------------------
__global__ void head_kernel(const float* __restrict__ t_out,
                            const float* __restrict__ g_out,
                            const float* __restrict__ fcW,
                            const float* __restrict__ fcb,
                            float* __restrict__ out) {
  const int row  = blockIdx.x;
  const int lane = threadIdx.x;
  float s = 0.0f;
#pragma unroll
  for (int i = 0; i < 4; ++i) {
    int c = lane * 4 + i;
    s += t_out[(size_t)row * ED + c] * fcW[c] +
         g_out[(size_t)row * ED + c] * fcW[ED + c];
  }
#pragma unroll
  for (int off = 1; off < 32; off <<= 1) s += __shfl_xor(s, off, 32);
  if (lane == 0) out[row] = 1.0f / (1.0f + __expf(-(s + fcb[0])));
}

// ---------------------------------------------------------------------------
extern "C" void kernel_launch(void* const* d_in, const int* in_sizes, int n_in,
                              void* d_out, int out_size, void* d_ws, size_t ws_size,
                              hipStream_t stream) {
  (void)in_sizes; (void)n_in; (void)out_size; (void)ws_size;
  const float* x       = (const float*)d_in[0];
  const float* t_Wqkv  = (const float*)d_in[1];
  const float* t_bqkv  = (const float*)d_in[2];
  const float* t_Wo    = (const float*)d_in[3];
  const float* t_bo    = (const float*)d_in[4];
  const float* t_ln1w  = (const float*)d_in[5];
  const float* t_ln1b  = (const float*)d_in[6];
  const float* t_W1    = (const float*)d_in[7];
  const float* t_b1    = (const float*)d_in[8];
  const float* t_W2    = (const float*)d_in[9];
  const float* t_b2    = (const float*)d_in[10];
  const float* t_ln2w  = (const float*)d_in[11];
  const float* t_ln2b  = (const float*)d_in[12];
  const float* g_W     = (const float*)d_in[13];
  const float* g_b     = (const float*)d_in[14];
  const float* fc_W    = (const float*)d_in[15];
  const float* fc_b    = (const float*)d_in[16];
  const int*   edges   = (const int*)d_in[17];
  float* out = (float*)d_out;

  // Workspace layout (floats)
  float* ws   = (float*)d_ws;
  float* xb   = ws;                          // 8192*128
  float* qkv  = xb  + (size_t)MROWS * ED;    // 8192*384
  float* obuf = qkv + (size_t)MROWS * 3 * ED;
  float* tmp  = obuf + (size_t)MROWS * ED;
  float* hid  = tmp  + (size_t)MROWS * ED;
  float* gha  = hid  + (size_t)MROWS * ED;
  float* ghb  = gha  + (size_t)MROWS * ED;
  float* gxw  = ghb  + (size_t)MROWS * ED;
  float* dinv = gxw  + (size_t)MROWS * ED;   // B*N

  const size_t actBytes = (size_t)MROWS * ED * sizeof(float);
  hipMemcpyAsync(xb,  x, actBytes, hipMemcpyDeviceToDevice, stream);
  hipMemcpyAsync(gha, x, actBytes, hipMemcpyDeviceToDevice, stream);

  // Degree / normalization (edges fixed across layers)
  zero_kernel<<<(BB * NN + 255) / 256, 256, 0, stream>>>(dinv, BB * NN);
  deg_kernel<<<dim3(NEDGE / 256, BB), 256, 0, stream>>>(edges, dinv);
  dinv_kernel<<<(BB * NN) / 256, 256, 0, stream>>>(dinv);

  const dim3 g384(MROWS / 16, (3 * ED) / 16);
  const dim3 g128(MROWS / 16, ED / 16);

  // Transformer encoder (8 layers)
  for (int l = 0; l < LT; ++l) {
    gemm_bias_kernel<0><<<g384, 32, 0, stream>>>(
        xb, t_Wqkv + (size_t)l * ED * 3 * ED, t_bqkv + l * 3 * ED, qkv, ED, 3 * ED);
    attention_kernel<<<dim3(NN / 16, HH, BB), 32, 0, stream>>>(qkv, obuf);
    gemm_bias_kernel<0><<<g128, 32, 0, stream>>>(
        obuf, t_Wo + (size_t)l * ED * ED, t_bo + l * ED, tmp, ED, ED);
    ln_residual_kernel<<<MROWS, 32, 0, stream>>>(xb, tmp, t_ln1w + l * ED, t_ln1b + l * ED);
    gemm_bias_kernel<1><<<g128, 32, 0, stream>>>(
        xb, t_W1 + (size_t)l * ED * ED, t_b1 + l * ED, hid, ED, ED);
    gemm_bias_kernel<0><<<g128, 32, 0, stream>>>(
        hid, t_W2 + (size_t)l * ED * ED, t_b2 + l * ED, tmp, ED, ED);
    ln_residual_kernel<<<MROWS, 32, 0, stream>>>(xb, tmp, t_ln2w + l * ED, t_ln2b + l * ED);
  }

  // GCN stack (4 layers), ping-pong
  float* ha = gha;
  float* hb = ghb;
  for (int l = 0; l < LG; ++l) {
    gemm_bias_kernel<0><<<g128, 32, 0, stream>>>(
        ha, g_W + (size_t)l * ED * ED, nullptr, gxw, ED, ED);
    gcn_init_kernel<<<dim3(NN, BB), ED, 0, stream>>>(dinv, gxw, g_b + l * ED, hb);
    gcn_scatter_kernel<<<dim3((NEDGE + 1) / 2, BB), 256, 0, stream>>>(edges, dinv, gxw, hb);
    float* t2 = ha; ha = hb; hb = t2;
  }

  // Final head: sigmoid(concat(t_out, g_out) @ fc_W + fc_b)
  head_kernel<<<MROWS, 32, 0, stream>>>(xb, ha, fc_W, fc_b, out);
}